// EdgeAttentionGAT_79637283602852
// MI455X (gfx1250) — compile-verified
//
#include <hip/hip_runtime.h>
#include <math.h>

// ---------------------------------------------------------------------------
// EdgeAttentionGAT on MI455X (gfx1250, wave32).
// Restructured dataflow:
//   temp[v] = deg(v)*h_proj[v]  (+)  sum_{edges at v} ef        (no (E,131) scatter)
//   h[n]    = deg[n]*(node_feats[n] @ Wfused + bfused) + efsum[n] @ Wg3
// where Wfused = Wp @ Wg[:128,:]  (computed on device).
// The N x 128 x 64 fp32 GEMM runs on V_WMMA_F32_16X16X4_F32.
// ---------------------------------------------------------------------------

#define IN_F   128
#define HOUT   64      // H * OUT
#define HEADS  4
#define OUTF   16
#define SLOPE  0.2f

typedef __attribute__((ext_vector_type(2))) float v2f;
typedef __attribute__((ext_vector_type(8))) float v8f;

// Monotonic float <-> ordered-uint mapping for atomic max over signed floats.
__device__ __forceinline__ unsigned f2ord(float f) {
    unsigned u = __float_as_uint(f);
    return (u & 0x80000000u) ? ~u : (u | 0x80000000u);
}
__device__ __forceinline__ float ord2f(unsigned u) {
    unsigned b = (u & 0x80000000u) ? (u & 0x7fffffffu) : ~u;
    return __uint_as_float(b);
}

// K0: zero accumulators, init softmax max (ordered enc of "below any real"),
// and seed d_out with the broadcast output bias (aggregation adds into it).
__global__ void k_init(float* __restrict__ deg, float* __restrict__ efs,
                       unsigned* __restrict__ m_enc, float* __restrict__ s,
                       float* __restrict__ out, const float* __restrict__ bg, int N) {
    int idx = blockIdx.x * blockDim.x + threadIdx.x;
    if (idx < N * HOUT) out[idx] = bg[idx & (HOUT - 1)];
    if (idx < N)        deg[idx] = 0.f;
    if (idx < 3 * N)    efs[idx] = 0.f;
    if (idx < HEADS * N) { m_enc[idx] = 0u; s[idx] = 0.f; }
}

// K1: Wfused[i][j] = sum_k Wp[i][k]*Wg[k][j];  bfused[j] = sum_k bp[k]*Wg[k][j]
__global__ void k_fuse(const float* __restrict__ Wp, const float* __restrict__ bp,
                       const float* __restrict__ Wg,
                       float* __restrict__ Wf, float* __restrict__ bf) {
    int t = blockIdx.x * blockDim.x + threadIdx.x;
    if (t < IN_F * HOUT) {
        int i = t >> 6, j = t & (HOUT - 1);
        float acc = 0.f;
        #pragma unroll 8
        for (int k = 0; k < IN_F; ++k) acc += Wp[i * IN_F + k] * Wg[k * HOUT + j];
        Wf[t] = acc;
    } else if (t < IN_F * HOUT + HOUT) {
        int j = t - IN_F * HOUT;
        float acc = 0.f;
        for (int k = 0; k < IN_F; ++k) acc += bp[k] * Wg[k * HOUT + j];
        bf[j] = acc;
    }
}

// K2: tiny edge MLP (3->6 relu ->3) + scatter ef-sums and degree counts.
__global__ void k_edge_mlp(const float* __restrict__ ef_in,
                           const int* __restrict__ src, const int* __restrict__ dst,
                           const float* __restrict__ W1, const float* __restrict__ b1,
                           const float* __restrict__ W2, const float* __restrict__ b2,
                           float* __restrict__ deg, float* __restrict__ efs, int E) {
    int e = blockIdx.x * blockDim.x + threadIdx.x;
    if (e >= E) return;
    float x0 = ef_in[e * 3 + 0], x1 = ef_in[e * 3 + 1], x2 = ef_in[e * 3 + 2];
    float hid[6];
    #pragma unroll
    for (int j = 0; j < 6; ++j) {
        float v = x0 * W1[j] + x1 * W1[6 + j] + x2 * W1[12 + j] + b1[j];
        hid[j] = v > 0.f ? v : 0.f;
    }
    float o[3];
    #pragma unroll
    for (int k = 0; k < 3; ++k) {
        float v = b2[k];
        #pragma unroll
        for (int j = 0; j < 6; ++j) v += hid[j] * W2[j * 3 + k];
        o[k] = v;
    }
    int sN = src[e], dN = dst[e];
    #pragma unroll
    for (int k = 0; k < 3; ++k) {
        atomicAdd(&efs[sN * 3 + k], o[k]);
        atomicAdd(&efs[dN * 3 + k], o[k]);
    }
    atomicAdd(&deg[sN], 1.0f);
    atomicAdd(&deg[dN], 1.0f);
}

// K3: (N,128)x(128,64) fp32 GEMM via V_WMMA_F32_16X16X4_F32.
// Block = 128 threads (4 wave32), one 16-node tile; wave w owns head w's 16 cols.
// Epilogue fuses deg/efsum terms and attn_l/attn_r reductions (width-16 shfl).
__global__ void __launch_bounds__(128)
k_node_gemm(const float* __restrict__ nf, const float* __restrict__ Wf,
            const float* __restrict__ bf, const float* __restrict__ Wg,
            const float* __restrict__ deg, const float* __restrict__ efs,
            const float* __restrict__ attn_l, const float* __restrict__ attn_r,
            float* __restrict__ hws, float* __restrict__ el, float* __restrict__ er,
            int N) {
    __shared__ float sA[16][IN_F + 4];      // padded: avoid 64-bank conflicts
    __shared__ float sB[IN_F][HOUT + 4];
    const int t  = threadIdx.x;
    const int m0 = blockIdx.x * 16;

    for (int idx = t; idx < 16 * IN_F; idx += 128) {     // A tile (coalesced)
        int r = idx >> 7, k = idx & (IN_F - 1);
        int n = m0 + r;
        sA[r][k] = (n < N) ? nf[(long long)n * IN_F + k] : 0.f;
    }
    for (int idx = t; idx < IN_F * HOUT; idx += 128)     // fused weights
        sB[idx >> 6][idx & (HOUT - 1)] = Wf[idx];
    __syncthreads();

    const int wave = t >> 5, lane = t & 31;
    const int half = lane >> 4, rrow = lane & 15;
    const int col  = wave * OUTF + rrow;                 // 0..63, head = wave

    v8f acc = {};
    #pragma unroll
    for (int k0 = 0; k0 < IN_F; k0 += 4) {
        const int kk = k0 + half * 2;                    // A: v0={K0,K2}, v1={K1,K3}
        v2f a, b;
        a.x = sA[rrow][kk];  a.y = sA[rrow][kk + 1];
        b.x = sB[kk][col];   b.y = sB[kk + 1][col];
        acc = __builtin_amdgcn_wmma_f32_16x16x4_f32(
                  false, a, false, b, (short)0, acc, false, false);
    }

    // D layout: lanes 0-15 rows r, lanes 16-31 rows r+8, N = lane%16.
    const float w0 = Wg[(IN_F + 0) * HOUT + col];
    const float w1 = Wg[(IN_F + 1) * HOUT + col];
    const float w2 = Wg[(IN_F + 2) * HOUT + col];
    const float bb = bf[col];
    float hv[8];
    #pragma unroll
    for (int r = 0; r < 8; ++r) {
        int n  = m0 + r + half * 8;
        int nn = (n < N) ? n : 0;
        float dg = deg[nn];
        float e0 = efs[nn * 3 + 0], e1 = efs[nn * 3 + 1], e2 = efs[nn * 3 + 2];
        float v = dg * (acc[r] + bb) + e0 * w0 + e1 * w1 + e2 * w2;
        hv[r] = v;
        if (n < N) hws[(long long)n * HOUT + col] = v;
    }

    const float al = attn_l[wave * OUTF + rrow];
    const float ar = attn_r[wave * OUTF + rrow];
    #pragma unroll
    for (int r = 0; r < 8; ++r) {
        float vl = hv[r] * al, vr = hv[r] * ar;
        #pragma unroll
        for (int off = 8; off >= 1; off >>= 1) {
            vl += __shfl_xor(vl, off, 16);
            vr += __shfl_xor(vr, off, 16);
        }
        int n = m0 + r + half * 8;
        if (rrow == 0 && n < N) {
            el[n * HEADS + wave] = vl;
            er[n * HEADS + wave] = vr;
        }
    }
}

// K4: per (edge, head) attention logit + segment max over dst (ordered-uint atomic).
__global__ void k_att_logits(const int* __restrict__ src, const int* __restrict__ dst,
                             const float* __restrict__ el, const float* __restrict__ er,
                             float* __restrict__ exw, unsigned* __restrict__ m_enc, int E) {
    int t = blockIdx.x * blockDim.x + threadIdx.x;
    if (t >= E * HEADS) return;
    int e = t >> 2, h = t & 3;
    float v = el[src[e] * HEADS + h] + er[dst[e] * HEADS + h];
    v = v > 0.f ? v : SLOPE * v;
    exw[t] = v;
    atomicMax(&m_enc[dst[e] * HEADS + h], f2ord(v));
}

// K5: ex = exp(e - m[dst]); segment sum over dst.
__global__ void k_exp_sum(const int* __restrict__ dst, const unsigned* __restrict__ m_enc,
                          float* __restrict__ exw, float* __restrict__ s, int E) {
    int t = blockIdx.x * blockDim.x + threadIdx.x;
    if (t >= E * HEADS) return;
    int e = t >> 2, h = t & 3;
    int d = dst[e];
    float ex = expf(exw[t] - ord2f(m_enc[d * HEADS + h]));
    exw[t] = ex;
    atomicAdd(&s[d * HEADS + h], ex);
}

// K6: out[dst] += alpha * h[src]   (d_out pre-seeded with bias).
__global__ void k_aggregate(const int* __restrict__ src, const int* __restrict__ dst,
                            const float* __restrict__ exw, const float* __restrict__ s,
                            const float* __restrict__ hws, float* __restrict__ out,
                            long long total) {
    long long t = (long long)blockIdx.x * blockDim.x + threadIdx.x;
    if (t >= total) return;
    int e = (int)(t >> 6), c = (int)(t & 63), h = c >> 4;
    int sN = src[e], dN = dst[e];
    float alpha = exw[e * HEADS + h] / s[dN * HEADS + h];
    atomicAdd(&out[dN * HOUT + c], alpha * hws[(long long)sN * HOUT + c]);
}

extern "C" void kernel_launch(void* const* d_in, const int* in_sizes, int n_in,
                              void* d_out, int out_size, void* d_ws, size_t ws_size,
                              hipStream_t stream) {
    const float* nf     = (const float*)d_in[0];
    const float* efin   = (const float*)d_in[1];
    const int*   src    = (const int*)  d_in[2];
    const int*   dst    = (const int*)  d_in[3];
    const float* W1     = (const float*)d_in[4];
    const float* b1     = (const float*)d_in[5];
    const float* W2     = (const float*)d_in[6];
    const float* b2     = (const float*)d_in[7];
    const float* Wp     = (const float*)d_in[8];
    const float* bp     = (const float*)d_in[9];
    const float* Wg     = (const float*)d_in[10];
    const float* attn_l = (const float*)d_in[11];
    const float* attn_r = (const float*)d_in[12];
    const float* bg     = (const float*)d_in[13];

    const int N = in_sizes[0] / IN_F;
    const int E = in_sizes[2];

    float* p = (float*)d_ws;
    float*    Wf    = p;              p += IN_F * HOUT;
    float*    bf    = p;              p += HOUT;
    float*    deg   = p;              p += N;
    float*    efs   = p;              p += 3 * N;
    float*    hws   = p;              p += (size_t)N * HOUT;
    float*    el    = p;              p += HEADS * N;
    float*    er    = p;              p += HEADS * N;
    unsigned* m_enc = (unsigned*)p;   p += HEADS * N;
    float*    s     = p;              p += HEADS * N;
    float*    exw   = p;              p += (size_t)E * HEADS;
    float*    out   = (float*)d_out;

    const int thr = 256;
    k_init<<<((size_t)N * HOUT + thr - 1) / thr, thr, 0, stream>>>(deg, efs, m_enc, s, out, bg, N);
    k_fuse<<<(IN_F * HOUT + HOUT + thr - 1) / thr, thr, 0, stream>>>(Wp, bp, Wg, Wf, bf);
    k_edge_mlp<<<(E + thr - 1) / thr, thr, 0, stream>>>(efin, src, dst, W1, b1, W2, b2, deg, efs, E);
    k_node_gemm<<<(N + 15) / 16, 128, 0, stream>>>(nf, Wf, bf, Wg, deg, efs, attn_l, attn_r,
                                                   hws, el, er, N);
    k_att_logits<<<((size_t)E * HEADS + thr - 1) / thr, thr, 0, stream>>>(src, dst, el, er, exw, m_enc, E);
    k_exp_sum<<<((size_t)E * HEADS + thr - 1) / thr, thr, 0, stream>>>(dst, m_enc, exw, s, E);
    const long long tot = (long long)E * HOUT;
    k_aggregate<<<(tot + thr - 1) / thr, thr, 0, stream>>>(src, dst, exw, s, hws, out, tot);
}